// GATCustom_79748952752833
// MI455X (gfx1250) — compile-verified
//
#include <hip/hip_runtime.h>
#include <math.h>

// GAT fused kernel for MI455X (gfx1250), wave32, WMMA f32 16x16x4.
// Shapes: B=64 batches, N=128 nodes, F=64 in, O=64 out. All fp32.

typedef __attribute__((ext_vector_type(2))) float v2f;
typedef __attribute__((ext_vector_type(8))) float v8f;

#define NB    64
#define NN    128
#define NF    64
#define NO    64
#define EIS   66   // padded LDS stride for ei/ej (kills bank conflicts on row-strided reads)
#define ATTS  20   // padded LDS stride for 128x16 att tile (A-operand reads)

__global__ __launch_bounds__(512) void gat_fused_kernel(
    const float* __restrict__ x,    // (8192, 64)
    const float* __restrict__ W,    // (64, 64)
    const float* __restrict__ av,   // (64, 1)
    const float* __restrict__ Wl,   // (64, 128)
    float* __restrict__ out)        // (8192, 64)
{
    __shared__ float s_ei[NN * EIS];   // ei[b] : 128 x 64 (stride 66)
    __shared__ float s_ej[NN * EIS];   // ej[b]
    __shared__ float s_h [NN * NO];    // h[b]  : 128 x 64
    __shared__ float s_att[NN * ATTS]; // one 128 x 16 att j-tile
    __shared__ float s_a[NO];
    __shared__ float s_red[32 * 16];   // per-(segment, col) partial max / sum
    __shared__ float s_cmax[16];
    __shared__ float s_csum[16];

    const int b     = blockIdx.x;
    const int tid   = threadIdx.x;
    const int wave  = tid >> 5;     // 0..15
    const int lane  = tid & 31;
    const int l16   = lane & 15;
    const int khalf = lane >> 4;    // 0/1 -> K-pair select per ISA A/B layout

    if (tid < NO) s_a[tid] = av[tid];

    const float* xb = x + (size_t)b * NN * NF;

    // ---------------- Phase 1: ei, ej, h via WMMA f32 16x16x4 ----------------
    // 96 jobs = mat{0:ei,1:ej,2:h} x mt(8) x nt(4)
    for (int job = wave; job < 96; job += 16) {
        const int mat  = job % 3;
        const int tile = job / 3;
        const int mt   = tile >> 2;       // 0..7  (16-row block)
        const int nt   = tile & 3;        // 0..3  (16-col block)
        const int n    = nt * 16 + l16;   // output column

        // Branchless B-fragment addressing (avoids per-k EXEC masking):
        //   mat==2 (h) : B[kk][n] = W[kk*64 + n]           -> base W+n,           stride 64
        //   mat==0 (ei): B[kk][n] = Wl[n*128 + kk]         -> base Wl+n*128,      stride 1
        //   mat==1 (ej): B[kk][n] = Wl[n*128 + 64 + kk]    -> base Wl+n*128+64,   stride 1
        const bool ish = (mat == 2);
        const float* bp = ish ? (W + n)
                              : (Wl + n * (2 * NF) + (mat == 1 ? NF : 0));
        const int bs = ish ? NO : 1;

        v8f acc = {};
        const float* xrow = xb + (mt * 16 + l16) * NF + 2 * khalf;
        const int kk0 = 2 * khalf;
        for (int k = 0; k < NF; k += 4) {
            v2f afrag = *(const v2f*)(xrow + k);          // A[m][kk], A[m][kk+1]
            const int kk = k + kk0;
            v2f bfrag;
            bfrag.x = bp[kk * bs];
            bfrag.y = bp[(kk + 1) * bs];
            acc = __builtin_amdgcn_wmma_f32_16x16x4_f32(
                false, afrag, false, bfrag, (short)0, acc, false, false);
        }
        // D layout: VGPR r -> row r (lanes 0-15) / row 8+r (lanes 16-31)
        float* dst;
        int stride;
        if      (mat == 0) { dst = s_ei; stride = EIS; }
        else if (mat == 1) { dst = s_ej; stride = EIS; }
        else               { dst = s_h;  stride = NO;  }
        const int row0 = mt * 16 + 8 * khalf;
        #pragma unroll
        for (int r = 0; r < 8; ++r)
            dst[(row0 + r) * stride + n] = acc[r];
    }
    __syncthreads();

    // ---------------- Phase 2: att tile -> softmax -> h' accumulation -------
    // wave w owns output row-block mt = w/2 and col-blocks nt = (w&1)*2 + {0,1}
    const int mtw = wave >> 1;
    v8f hacc[2];
    hacc[0] = (v8f){};
    hacc[1] = (v8f){};

    const int jj  = tid & 15;       // column within j-tile
    const int seg = tid >> 4;       // 0..31 row segment (4 rows each)
    const int i0  = seg * 4;

    for (int jt = 0; jt < 8; ++jt) {
        // --- att[i][jj] = sum_o relu(ei[i][o] + ej[j][o]) * a[o]  (VALU; relu blocks matmul form)
        {
            const int j = jt * 16 + jj;
            const float* ejrow = s_ej + j * EIS;
            for (int ii = 0; ii < 4; ++ii) {
                const int i = i0 + ii;
                const float* eirow = s_ei + i * EIS;
                float s = 0.f;
                #pragma unroll 8
                for (int o = 0; o < NO; ++o) {
                    float v = eirow[o] + ejrow[o];
                    v = v > 0.f ? v : 0.f;
                    s = fmaf(v, s_a[o], s);
                }
                s_att[i * ATTS + jj] = s;
            }
        }
        __syncthreads();

        // --- softmax over i (per column jj): tree reduce max, then sum of exp
        {
            float m = -3.4e38f;
            for (int ii = 0; ii < 4; ++ii)
                m = fmaxf(m, s_att[(i0 + ii) * ATTS + jj]);
            s_red[seg * 16 + jj] = m;
        }
        __syncthreads();
        if (tid < 16) {
            float m = -3.4e38f;
            for (int sgi = 0; sgi < 32; ++sgi) m = fmaxf(m, s_red[sgi * 16 + tid]);
            s_cmax[tid] = m;
        }
        __syncthreads();
        {
            const float cm = s_cmax[jj];
            float s = 0.f;
            for (int ii = 0; ii < 4; ++ii) {
                float e = __expf(s_att[(i0 + ii) * ATTS + jj] - cm);
                s_att[(i0 + ii) * ATTS + jj] = e;
                s += e;
            }
            s_red[seg * 16 + jj] = s;
        }
        __syncthreads();
        if (tid < 16) {
            float s = 0.f;
            for (int sgi = 0; sgi < 32; ++sgi) s += s_red[sgi * 16 + tid];
            s_csum[tid] = 1.0f / s;
        }
        __syncthreads();
        {
            const float inv = s_csum[jj];
            for (int ii = 0; ii < 4; ++ii)
                s_att[(i0 + ii) * ATTS + jj] *= inv;
        }
        __syncthreads();

        // --- hacc += att_tile(M=16 rows mtw*16.., K=16) x h(rows jt*16.., 16 cols)
        #pragma unroll
        for (int q = 0; q < 2; ++q) {
            const int nt = (wave & 1) * 2 + q;
            const int n  = nt * 16 + l16;
            #pragma unroll
            for (int kk = 0; kk < 16; kk += 4) {
                const int krow = kk + 2 * khalf;
                v2f afrag = *(const v2f*)(s_att + (mtw * 16 + l16) * ATTS + krow);
                v2f bfrag;
                bfrag.x = s_h[(jt * 16 + krow) * NO + n];
                bfrag.y = s_h[(jt * 16 + krow + 1) * NO + n];
                hacc[q] = __builtin_amdgcn_wmma_f32_16x16x4_f32(
                    false, afrag, false, bfrag, (short)0, hacc[q], false, false);
            }
        }
        __syncthreads();   // att tile reused next jt
    }

    // ---------------- Epilogue: scatter C/D fragments to global --------------
    #pragma unroll
    for (int q = 0; q < 2; ++q) {
        const int nt   = (wave & 1) * 2 + q;
        const int n    = nt * 16 + l16;
        const int row0 = mtw * 16 + 8 * khalf;
        #pragma unroll
        for (int r = 0; r < 8; ++r)
            out[((size_t)b * NN + row0 + r) * NO + n] = hacc[q][r];
    }
}

extern "C" void kernel_launch(void* const* d_in, const int* in_sizes, int n_in,
                              void* d_out, int out_size, void* d_ws, size_t ws_size,
                              hipStream_t stream) {
    (void)in_sizes; (void)n_in; (void)out_size; (void)d_ws; (void)ws_size;
    const float* x  = (const float*)d_in[0];
    const float* W  = (const float*)d_in[1];
    const float* a  = (const float*)d_in[2];
    const float* Wl = (const float*)d_in[3];
    // d_in[4] = batch indices (int64); structure is known statically, unused.
    float* out = (float*)d_out;

    gat_fused_kernel<<<dim3(NB), dim3(512), 0, stream>>>(x, W, a, Wl, out);
}